// LSAALoss_39737037422868
// MI455X (gfx1250) — compile-verified
//
#include <hip/hip_runtime.h>
#include <math.h>

#define TAU 1.0f
#define NEG_BIG -1000000000.0f
#define NUM_ITERS 20
#define BB 8
#define SS 512
#define VV 32000
#define KPAD 31488.0f   // (V - S) identical padded columns

typedef float v2f __attribute__((ext_vector_type(2)));
typedef float v8f __attribute__((ext_vector_type(8)));

// Exact full-wave (32-lane) sum using V_WMMA_F32_16X16X4_F32.
// A = 16x4 with each lane contributing {v, 0}; B = ones(4x16); C = 0.
// D[m,n] = sum_k A[m,k]  -> summing D's 8 VGPRs gives rows 0-7 in lanes 0-15
// and rows 8-15 in lanes 16-31; one xor-16 shuffle completes the total.
__device__ __forceinline__ float wave_sum_wmma(float v) {
  v2f a; a[0] = v;    a[1] = 0.0f;
  v2f b; b[0] = 1.0f; b[1] = 1.0f;
  v8f c = {};
  v8f d = __builtin_amdgcn_wmma_f32_16x16x4_f32(
      /*neg_a=*/false, a, /*neg_b=*/false, b,
      /*c_mod=*/(short)0, c, /*reuse_a=*/false, /*reuse_b=*/false);
  float r = d[0] + d[1] + d[2] + d[3] + d[4] + d[5] + d[6] + d[7];
  r += __shfl_xor(r, 16, 32);
  return r;  // full wave sum, identical in every lane
}

// ---------------- Kernel 1: ce[b,s] = logsumexp_V(pred[b,s,:]) - pred[b,s,t] --
__global__ __launch_bounds__(256) void ce_kernel(const float* __restrict__ pred,
                                                 const int* __restrict__ tgt,
                                                 float* __restrict__ ce) {
  const int row = blockIdx.x;  // 0 .. B*S-1
  const float4* p4 = reinterpret_cast<const float4*>(pred + (size_t)row * VV);
  const int n4 = VV / 4;  // 8000

  float m = -INFINITY, s = 0.0f;
#pragma unroll 2
  for (int i = threadIdx.x; i < n4; i += 256) {
    float4 v = p4[i];
    float cmx = fmaxf(fmaxf(v.x, v.y), fmaxf(v.z, v.w));
    float nm = fmaxf(m, cmx);
    s = s * __expf(m - nm) + __expf(v.x - nm) + __expf(v.y - nm) +
        __expf(v.z - nm) + __expf(v.w - nm);
    m = nm;
  }

  __shared__ float smax[8];
  __shared__ float ssum[8];
  const int w = threadIdx.x >> 5;

  // block max
  float wm = m;
  for (int off = 16; off; off >>= 1) wm = fmaxf(wm, __shfl_xor(wm, off, 32));
  if ((threadIdx.x & 31) == 0) smax[w] = wm;
  __syncthreads();
  float M = smax[0];
  for (int i = 1; i < 8; i++) M = fmaxf(M, smax[i]);

  // rescale and block sum (WMMA wave sums + LDS combine)
  float sv = s * __expf(m - M);
  float wsum = wave_sum_wmma(sv);
  if ((threadIdx.x & 31) == 0) ssum[w] = wsum;
  __syncthreads();

  if (threadIdx.x == 0) {
    float Stot = 0.0f;
    for (int i = 0; i < 8; i++) Stot += ssum[i];
    float lse = M + logf(Stot);
    int t = tgt[row];
    float tv = pred[(size_t)row * VV + (size_t)t];
    ce[row] = lse - tv;
  }
}

// ---------------- Block reductions over 512 threads (16 waves) ---------------
__device__ __forceinline__ float block_max512(float v, float* red) {
  for (int off = 16; off; off >>= 1) v = fmaxf(v, __shfl_xor(v, off, 32));
  const int w = threadIdx.x >> 5;
  __syncthreads();
  if ((threadIdx.x & 31) == 0) red[w] = v;
  __syncthreads();
  float r = red[0];
  for (int i = 1; i < 16; i++) r = fmaxf(r, red[i]);
  return r;
}

__device__ __forceinline__ float block_sum512(float v, float* red) {
  float ws = wave_sum_wmma(v);
  const int w = threadIdx.x >> 5;
  __syncthreads();
  if ((threadIdx.x & 31) == 0) red[w] = ws;
  __syncthreads();
  float r = 0.0f;
  for (int i = 0; i < 16; i++) r += red[i];
  return r;
}

__device__ __forceinline__ float block_lse512(float v, float* red) {
  float M = block_max512(v, red);
  float e = __expf(v - M);
  float Ssum = block_sum512(e, red);
  return M + logf(Ssum);
}

// ---------------- Kernel 2: collapsed Sinkhorn + loss per batch --------------
// State per (b,s): a = cost column 0, d = any of the 31488 identical padded
// columns. Row-lse over the 31489-wide row is m + log(e^{a-m} + K e^{d-m}).
__global__ __launch_bounds__(512) void sinkhorn_kernel(const float* __restrict__ ce,
                                                       float* __restrict__ partial) {
  const int b = blockIdx.x;
  __shared__ float red[16];

  const float myce = ce[b * SS + threadIdx.x];
  float a = -myce / TAU;
  float d = NEG_BIG;

  for (int it = 0; it < NUM_ITERS; ++it) {
    // axis=2 (row) normalization — elementwise
    float m = fmaxf(a, d);
    float r = m + logf(__expf(a - m) + KPAD * __expf(d - m));
    a -= r;
    d -= r;
    // axis=1 (column) normalization — over the 512 threads
    a -= block_lse512(a, red);
    d -= block_lse512(d, red);
  }

  // loss_b = sum_s ce[s] * (e^{a[s]} + K e^{d[s]})
  float p = __expf(a) + KPAD * __expf(d);
  float tot = block_sum512(myce * p, red);
  if (threadIdx.x == 0) partial[b] = tot;
}

// ---------------- Kernel 3: mean over batches --------------------------------
__global__ void finalize_kernel(const float* __restrict__ partial,
                                float* __restrict__ out) {
  if (threadIdx.x == 0 && blockIdx.x == 0) {
    float t = 0.0f;
    for (int i = 0; i < BB; i++) t += partial[i];
    out[0] = t / (float)BB;
  }
}

extern "C" void kernel_launch(void* const* d_in, const int* in_sizes, int n_in,
                              void* d_out, int out_size, void* d_ws, size_t ws_size,
                              hipStream_t stream) {
  const float* pred = (const float*)d_in[0];
  const int* tgt = (const int*)d_in[1];   // targets (int32 per harness convention)
  float* ce = (float*)d_ws;               // B*S floats
  float* partial = ce + BB * SS;          // B floats
  float* out = (float*)d_out;

  ce_kernel<<<BB * SS, 256, 0, stream>>>(pred, tgt, ce);
  sinkhorn_kernel<<<BB, SS, 0, stream>>>(ce, partial);
  finalize_kernel<<<1, 32, 0, stream>>>(partial, out);
}